// CoreArchitect_42631845380381
// MI455X (gfx1250) — compile-verified
//
#include <hip/hip_runtime.h>

// ---------------------------------------------------------------------------
// Capsule dynamic routing for MI455X (gfx1250, wave32).
//
// Shapes: N=32, C=32, D=16 (P=4), W=H=16 -> WH=256, K=10, M=C*WH=8192.
// V (N,M,K,D) = 160 MB: materialized once in d_ws, then L2-resident
// (192 MB L2) for the 3 routing passes. HBM floor ~180 MB => ~8 us.
//
// Kernel 1 maps the einsum onto V_WMMA_F32_16X16X4_F32:
//   per wave: B(4x16) = weights for 4 classes x 4 'l' values (loaded ONCE),
//             then loop 8 wh-tiles: A(16x4) = lr rows for 4 spatial positions
//             x 4 'i' values, contraction j=4 -> 8 WMMAs per wave.
// ---------------------------------------------------------------------------

typedef float v2f __attribute__((ext_vector_type(2)));
typedef float v8f __attribute__((ext_vector_type(8)));

#define NBATCH 32
#define CCH    32
#define DCAP   16
#define WHDIM  256
#define KCLS   10
#define MDIM   (CCH * WHDIM)   // 8192
#define TILES_PER_WAVE 8       // wh-tiles (of 4 positions) per wave

// ------------------------- kernel 1: V via WMMA ----------------------------
// wave -> (n, c, whGroup of 8 tiles, kgroup); 32*32*8*3 = 24576 waves.
__global__ __launch_bounds__(256) void k_compute_V(
    const float* __restrict__ l,       // (N, C, D, WH)
    const float* __restrict__ weight,  // (C, K, 4, 4)
    float* __restrict__ V)             // (N, M, K, D)
{
  const int lane = threadIdx.x & 31;
  int wid = blockIdx.x * 8 + (threadIdx.x >> 5);

  const int kg  = wid % 3;  wid /= 3;
  const int whG = wid & 7;  wid >>= 3;   // group of 8 wh-tiles
  const int c   = wid & 31; wid >>= 5;
  const int n   = wid;                   // 0..31
  const int k0  = kg * 4;

  // ---- B (4x16 f32), loaded once per wave ---------------------------------
  // Rows striped across lanes: lanes 0-15 = rows j0=0/1, lanes 16-31 = 2/3
  // (VGPR0 = row j0, VGPR1 = row j0+1). Column col = 4*kk + ll.
  const int col = lane & 15;
  const int kk  = col >> 2;
  const int ll  = col & 3;
  const int k   = k0 + kk;
  const int kc  = (k < KCLS) ? k : 0;                 // safe index, no branch
  const float mask = (k < KCLS) ? 1.0f : 0.0f;        // zero padded classes
  const int j0  = (lane >> 4) << 1;                   // 0 or 2
  const float* wbase = weight + ((long)c * KCLS + kc) * 16;
  v2f B;
  B.x = mask * wbase[ j0      * 4 + ll];              // w[c,k,j0,ll]
  B.y = mask * wbase[(j0 + 1) * 4 + ll];              // w[c,k,j0+1,ll]

  // ---- A row mapping (32-bit 16x4, ISA 7.12.2) ----------------------------
  // Row m = lane&15 -> (wh_local = m>>2, i = m&3); lanes 0-15 hold j=0/1 in
  // VGPR0/1, lanes 16-31 hold j=2/3.
  const int m   = lane & 15;
  const int whl = m >> 2;
  const int ii  = m & 3;
  const float* lrow =
      l + ((long)(n * CCH + c) * DCAP + 4 * ii + j0) * WHDIM;

  const int hi = (lane >> 4) * 8;
  const long vrow0 =
      (((long)n * MDIM + (long)c * WHDIM) * KCLS + k) * DCAP + (4 * (hi ? 0 : 0));

#pragma unroll
  for (int tile = 0; tile < TILES_PER_WAVE; ++tile) {
    const int wh_base = (whG * TILES_PER_WAVE + tile) * 4;

    v2f A;
    A.x = lrow[wh_base + whl];            // lr[n,c,wh,i,j0]
    A.y = lrow[WHDIM + wh_base + whl];    // lr[n,c,wh,i,j0+1]

    v8f Cacc = {};
    // (neg_a, A, neg_b, B, c_mod, C, reuse_a, reuse_b)
    Cacc = __builtin_amdgcn_wmma_f32_16x16x4_f32(
        false, A, false, B, (short)0, Cacc, false, false);

    // C/D layout: VGPR r holds row (r + 8*(lane>=16)), column = lane&15.
    if (k < KCLS) {
#pragma unroll
      for (int r = 0; r < 8; ++r) {
        const int mr   = r + hi;          // output row 0..15
        const int whl2 = mr >> 2;
        const int i2   = mr & 3;
        const long vidx = vrow0 + (long)(wh_base + whl2) * (KCLS * DCAP) +
                          (4 * i2 + ll);
        V[vidx] = Cacc[r];
      }
    }
    // exec re-converges here before the next WMMA iteration
  }
  (void)vrow0;
}

// ---------------- kernel 2: b = <gv, V> (init) or b += (update) ------------
__global__ __launch_bounds__(256) void k_dot_b(
    const float* __restrict__ V, const float* __restrict__ gv,
    float* __restrict__ b, int add)
{
  const long idx = (long)blockIdx.x * blockDim.x + threadIdx.x; // (n*M+m)*K+k
  if (idx >= (long)NBATCH * MDIM * KCLS) return;
  const int  k  = (int)(idx % KCLS);
  const long nm = idx / KCLS;
  const int  n  = (int)(nm / MDIM);
  const float* vp = V + idx * DCAP;
  const float* gp = gv + ((long)n * KCLS + k) * DCAP;
  float acc = 0.0f;
#pragma unroll
  for (int d = 0; d < DCAP; ++d) acc += vp[d] * gp[d];
  if (add) b[idx] += acc; else b[idx] = acc;
}

// ---------------------------- zero s ---------------------------------------
__global__ void k_zero(float* __restrict__ p, int n) {
  int i = blockIdx.x * blockDim.x + threadIdx.x;
  if (i < n) p[i] = 0.0f;
}

// -------- kernel 3: softmax over K + weighted reduction over M -------------
// block = (256 m's of one n); phase 1: softmax into LDS; phase 2: 160 threads
// (k,d) each accumulate over 256 m's (coalesced 160-float rows of V), then
// one atomicAdd per (n,k,d) per block.
__global__ __launch_bounds__(256) void k_route(
    const float* __restrict__ V, const float* __restrict__ b,
    float* __restrict__ s)
{
  __shared__ float csm[256][KCLS];
  const int n     = blockIdx.y;
  const int mbase = blockIdx.x * 256;
  const int t     = threadIdx.x;

  {
    const float* bp = b + ((long)n * MDIM + mbase + t) * KCLS;
    float bv[KCLS], mx = -1e30f;
#pragma unroll
    for (int k = 0; k < KCLS; ++k) { bv[k] = bp[k]; mx = fmaxf(mx, bv[k]); }
    float sum = 0.0f;
#pragma unroll
    for (int k = 0; k < KCLS; ++k) { bv[k] = __expf(bv[k] - mx); sum += bv[k]; }
    const float inv = 1.0f / sum;
#pragma unroll
    for (int k = 0; k < KCLS; ++k) csm[t][k] = bv[k] * inv;
  }
  __syncthreads();

  if (t < KCLS * DCAP) {
    const int k = t / DCAP, d = t % DCAP;
    float acc = 0.0f;
    const float* vp = V + (((long)n * MDIM + mbase) * KCLS + k) * DCAP + d;
    for (int mm = 0; mm < 256; ++mm)
      acc += csm[mm][k] * vp[(long)mm * KCLS * DCAP];
    atomicAdd(&s[(n * KCLS + k) * DCAP + d], acc);
  }
}

// -------------------- kernel 4: squash (+ final a) -------------------------
__global__ void k_squash(const float* __restrict__ s, float* __restrict__ v,
                         float* __restrict__ a_out, int last)
{
  const int idx = blockIdx.x * blockDim.x + threadIdx.x;  // n*K + k
  if (idx >= NBATCH * KCLS) return;
  const float* sp = s + idx * DCAP;
  float sq = 0.0f;
#pragma unroll
  for (int d = 0; d < DCAP; ++d) sq += sp[d] * sp[d];
  const float scale = sq / (1.0f + sq);
  const float inv   = scale * rsqrtf(sq);
#pragma unroll
  for (int d = 0; d < DCAP; ++d) v[idx * DCAP + d] = sp[d] * inv;
  if (last) {
    float nq = 0.0f;
#pragma unroll
    for (int d = 0; d < DCAP; ++d) {
      const float vd = sp[d] * inv;
      nq += vd * vd;
    }
    const float nrm = sqrtf(nq);
    a_out[idx] = 1.0f / (1.0f + __expf(-nrm));
  }
}

// ---------------------------------------------------------------------------
extern "C" void kernel_launch(void* const* d_in, const int* in_sizes, int n_in,
                              void* d_out, int out_size, void* d_ws,
                              size_t ws_size, hipStream_t stream)
{
  (void)in_sizes; (void)n_in; (void)out_size; (void)ws_size;
  const float* l = (const float*)d_in[0];       // (32,32,16,16,16)
  const float* g = (const float*)d_in[1];       // (32,10,16)
  const float* w = (const float*)d_in[2];       // (1,32,1,10,4,4)
  float* out = (float*)d_out;                   // a:320 then v:5120

  // workspace layout (floats)
  float* V    = (float*)d_ws;                              // 41,943,040
  float* b    = V + (long)NBATCH * MDIM * KCLS * DCAP;     //  2,621,440
  float* s    = b + (long)NBATCH * MDIM * KCLS;            //      5,120
  float* vtmp = s + NBATCH * KCLS * DCAP;                  //      5,120

  // 1) V via WMMA: 24576 waves (8 WMMAs each) -> 3072 blocks of 8 waves.
  k_compute_V<<<3072, 256, 0, stream>>>(l, w, V);

  // 2) initial logits b
  const int nbk = (NBATCH * MDIM * KCLS + 255) / 256;      // 10240
  k_dot_b<<<nbk, 256, 0, stream>>>(V, g, b, 0);

  // 3) routing iterations
  for (int it = 0; it < 3; ++it) {
    const int last = (it == 2);
    k_zero<<<(NBATCH * KCLS * DCAP + 255) / 256, 256, 0, stream>>>(
        s, NBATCH * KCLS * DCAP);
    dim3 grid(MDIM / 256, NBATCH);
    k_route<<<grid, 256, 0, stream>>>(V, b, s);
    float* vdst = last ? (out + NBATCH * KCLS) : vtmp;
    k_squash<<<(NBATCH * KCLS + 31) / 32, 32, 0, stream>>>(s, vdst, out, last);
    if (!last) k_dot_b<<<nbk, 256, 0, stream>>>(V, vtmp, b, 1);
  }
}